// Coords2Elec_7730941132977
// MI455X (gfx1250) — compile-verified
//
#include <hip/hip_runtime.h>
#include <math.h>

#define BOX 128
#define BOX3 (BOX * BOX * BOX)
#define WR 8
#define WDIM (2 * WR + 1)
#define WPTS (WDIM * WDIM * WDIM)
#define NITER 30
#define XCH 16   // x-planes marched per Jacobi block

static constexpr float RES        = 1.0f;
static constexpr float EPS_IN     = 6.5f;
static constexpr float EPS_OUT    = 79.0f;
static constexpr float ION_SIZE   = 1.0f;
static constexpr float WAT_SIZE   = 1.4f;
static constexpr float ASIGMA     = 2.0f;
static constexpr float KAPPA02    = 0.106f;
static constexpr float CHARGE_CONV = 7046.52f;

// CDNA5 async global->LDS path (device pass only; host pass falls back).
#if defined(__HIP_DEVICE_COMPILE__) && defined(__gfx1250__) && \
    __has_builtin(__builtin_amdgcn_global_load_async_to_lds_b32) && \
    __has_builtin(__builtin_amdgcn_s_wait_asynccnt)
#define USE_ASYNC_LDS 1
typedef __attribute__((address_space(1))) int gas_int;   // global (addrspace 1)
typedef __attribute__((address_space(3))) int lds_int;   // LDS    (addrspace 3)
#else
#define USE_ASYNC_LDS 0
#endif

// ---------------------------------------------------------------------------
// 1) eps channel accumulation: grid (N, 4, B) -> zero index divisions.
//    One block per (atom, channel, batch); 4913 window points / 256 threads.
//    Transcendental bound; atomics land in the L2-resident eps slice of d_out.
// ---------------------------------------------------------------------------
__global__ void __launch_bounds__(256)
eps_accum_kernel(const float* __restrict__ coords,
                 const float* __restrict__ params,
                 const int*   __restrict__ num_atoms,
                 float*       __restrict__ acc,   // [B][4][BOX3]
                 int N)
{
    const int a = blockIdx.x;
    const int c = blockIdx.y;
    const int b = blockIdx.z;
    if (a >= num_atoms[b]) return;   // atom_mask: masked entries contribute 0

    const float offx = (c == 0) ? 0.5f : 0.0f;
    const float offy = (c == 1) ? 0.5f : 0.0f;
    const float offz = (c == 2) ? 0.5f : 0.0f;
    const float extra = (c == 3) ? ION_SIZE : WAT_SIZE;

    const float x = coords[(size_t)b * 3 * N + a * 3 + 0];
    const float y = coords[(size_t)b * 3 * N + a * 3 + 1];
    const float z = coords[(size_t)b * 3 * N + a * 3 + 2];
    const float thr = params[((size_t)b * N + a) * 2 + 1] + extra;

    // jnp.round == round-half-even == rintf
    const int ix0 = (int)rintf(x / RES - offx);
    const int iy0 = (int)rintf(y / RES - offy);
    const int iz0 = (int)rintf(z / RES - offz);

    // dx = (ix + offx)*RES - x  ->  (float)ix + cx  (RES == 1)
    const float cx = offx * RES - x;
    const float cy = offy * RES - y;
    const float cz = offz * RES - z;

    float* __restrict__ accc = acc + (size_t)(b * 4 + c) * BOX3;

    for (int t = threadIdx.x; t < WPTS; t += 256) {
        const int oz = t % WDIM;                 // WDIM is a literal: mul-shift
        const int oy = (t / WDIM) % WDIM;
        const int ox = t / (WDIM * WDIM);
        const int ix = ix0 + ox - WR;
        const int iy = iy0 + oy - WR;
        const int iz = iz0 + oz - WR;
        if ((unsigned)ix < BOX && (unsigned)iy < BOX && (unsigned)iz < BOX) {
            const float dx = (float)ix * RES + cx;
            const float dy = (float)iy * RES + cy;
            const float dz = (float)iz * RES + cz;
            const float r = sqrtf(dx * dx + dy * dy + dz * dz + 1e-12f);
            float rho = 0.5f * (1.0f - erff((r - thr) / ASIGMA));
            rho = fminf(fmaxf(rho, 0.0f), 1.0f - 1e-6f);
            const float logv = log1pf(-rho);
            atomicAdd(accc + ((ix * BOX + iy) * BOX + iz), logv);
        }
    }
}

// ---------------------------------------------------------------------------
// 2) eps finalize (in place), float4-vectorized (b128 load/store).
//    rho = exp(acc); ch 0..2 -> (1-rho)*(ei-eo)+eo, ch 3 -> 1-rho.
//    BOX3 % 4 == 0 so a float4 never straddles a channel boundary.
// ---------------------------------------------------------------------------
__global__ void eps_finalize_kernel(float4* __restrict__ eps4, long long n4)
{
    const long long i = (long long)blockIdx.x * blockDim.x + threadIdx.x;
    if (i >= n4) return;
    const int c = (int)((i / (BOX3 / 4)) & 3);   // power of 2: shift
    float4 v = eps4[i];
    if (c < 3) {
        v.x = (1.0f - expf(v.x)) * (EPS_IN - EPS_OUT) + EPS_OUT;
        v.y = (1.0f - expf(v.y)) * (EPS_IN - EPS_OUT) + EPS_OUT;
        v.z = (1.0f - expf(v.z)) * (EPS_IN - EPS_OUT) + EPS_OUT;
        v.w = (1.0f - expf(v.w)) * (EPS_IN - EPS_OUT) + EPS_OUT;
    } else {
        v.x = 1.0f - expf(v.x);
        v.y = 1.0f - expf(v.y);
        v.z = 1.0f - expf(v.z);
        v.w = 1.0f - expf(v.w);
    }
    eps4[i] = v;
}

// ---------------------------------------------------------------------------
// 3) trilinear charge deposition, grid (ceil(N/256), B) -> no divisions.
//    CHARGE_CONV folded into the deposited value.
// ---------------------------------------------------------------------------
__global__ void q_scatter_kernel(const float* __restrict__ coords,
                                 const float* __restrict__ params,
                                 const int*   __restrict__ num_atoms,
                                 float*       __restrict__ q,
                                 int N)
{
    const int a = blockIdx.x * blockDim.x + threadIdx.x;
    const int b = blockIdx.y;
    if (a >= N || a >= num_atoms[b]) return;

    const float gx = coords[(size_t)b * 3 * N + a * 3 + 0] / RES;
    const float gy = coords[(size_t)b * 3 * N + a * 3 + 1] / RES;
    const float gz = coords[(size_t)b * 3 * N + a * 3 + 2] / RES;
    const float ch = params[((size_t)b * N + a) * 2 + 0] * CHARGE_CONV;

    const float fx0 = floorf(gx), fy0 = floorf(gy), fz0 = floorf(gz);
    const int i0 = (int)fx0, j0 = (int)fy0, k0 = (int)fz0;
    const float fx = gx - fx0, fy = gy - fy0, fz = gz - fz0;

    float* __restrict__ qb = q + (size_t)b * BOX3;
#pragma unroll
    for (int cx = 0; cx < 2; ++cx)
#pragma unroll
        for (int cy = 0; cy < 2; ++cy)
#pragma unroll
            for (int cz = 0; cz < 2; ++cz) {
                const int ix = i0 + cx, iy = j0 + cy, iz = k0 + cz;
                if ((unsigned)ix < BOX && (unsigned)iy < BOX && (unsigned)iz < BOX) {
                    const float w = (cx ? fx : 1.0f - fx) *
                                    (cy ? fy : 1.0f - fy) *
                                    (cz ? fz : 1.0f - fz);
                    atomicAdd(qb + ((ix * BOX + iy) * BOX + iz), w * ch);
                }
            }
}

// ---------------------------------------------------------------------------
// 4) Jacobi sweep, x-marching pencil version.
//    Block = one (b,y) z-line (128 threads = 4 wave32s) walking XCH x-planes.
//    The phi center line is double-buffered in LDS via the CDNA5 async
//    global->LDS DMA: the DMA for plane x+1 is issued BEFORE the eps/q/phi-y
//    global loads and FMA chain for plane x, and waited only when the x+1
//    value is actually needed -> DMA overlaps compute. z±1 come from ds_load,
//    ex[x-1] rolls in a register. Everything is L2-resident (<192MB).
// ---------------------------------------------------------------------------
__global__ void __launch_bounds__(BOX)
jacobi_kernel(const float* __restrict__ phi_in,
              float*       __restrict__ phi_out,
              const float* __restrict__ eps,   // [B][4][BOX3]
              const float* __restrict__ q)     // [B][BOX3]
{
    const int tz = threadIdx.x;          // 0..127 (z, contiguous)
    const int x0 = blockIdx.x * XCH;
    const int y  = blockIdx.y;
    const int b  = blockIdx.z;

    __shared__ float sbuf[2][BOX + 2];   // double-buffered phi z-lines + halo

    const size_t pb = (size_t)b * BOX3;
    const float* __restrict__ phib = phi_in + pb;
    const float* __restrict__ ex = eps + (size_t)(b * 4 + 0) * BOX3;
    const float* __restrict__ ey = eps + (size_t)(b * 4 + 1) * BOX3;
    const float* __restrict__ ez = eps + (size_t)(b * 4 + 2) * BOX3;
    const float* __restrict__ lm = eps + (size_t)(b * 4 + 3) * BOX3;
    const float* __restrict__ qb = q + pb;

    if (tz == 0) {
        sbuf[0][0] = 0.0f; sbuf[0][BOX + 1] = 0.0f;
        sbuf[1][0] = 0.0f; sbuf[1][BOX + 1] = 0.0f;
    }

    int s = (x0 * BOX + y) * BOX + tz;

    // preload plane x0's line into sbuf[x0&1]
#if USE_ASYNC_LDS
    __builtin_amdgcn_global_load_async_to_lds_b32(
        (gas_int*)(phib + s), (lds_int*)&sbuf[x0 & 1][tz + 1], 0, 0);
    __builtin_amdgcn_s_wait_asynccnt(0);
#else
    sbuf[x0 & 1][tz + 1] = phib[s];
#endif
    float p_xm = (x0 > 0) ? phib[s - BOX * BOX] : 0.0f;
    float exm  = (x0 > 0) ? ex[s - BOX * BOX]   : 0.0f;
    __syncthreads();

    for (int x = x0; x < x0 + XCH; ++x) {
        const int cur = x & 1, nxt = cur ^ 1;
        const bool has_xp = (x + 1 < BOX);

        // kick off the DMA for plane x+1 while plane x is computed
#if USE_ASYNC_LDS
        if (has_xp)
            __builtin_amdgcn_global_load_async_to_lds_b32(
                (gas_int*)(phib + s + BOX * BOX),
                (lds_int*)&sbuf[nxt][tz + 1], 0, 0);
#else
        if (has_xp) sbuf[nxt][tz + 1] = phib[s + BOX * BOX];
#endif
        if (x + 2 < BOX)   // speculative L2 warm-up two planes ahead
            __builtin_prefetch(phib + s + 2 * BOX * BOX, 0, 0);

        const float p_c = sbuf[cur][tz + 1];
        const float pzp = sbuf[cur][tz + 2];
        const float pzm = sbuf[cur][tz];

        const float ex_c = ex[s];
        const float ey_c = ey[s];
        const float ey_m = (y > 0)  ? ey[s - BOX] : 0.0f;
        const float ez_c = ez[s];
        const float ez_m = (tz > 0) ? ez[s - 1]   : 0.0f;
        const float lam  = lm[s];
        const float pyp  = (y + 1 < BOX) ? phib[s + BOX] : 0.0f;
        const float pym  = (y > 0)       ? phib[s - BOX] : 0.0f;
        const float rhs  = qb[s] * (RES * RES);

        float num = exm * p_xm + ey_c * pyp + ey_m * pym +
                    ez_c * pzp + ez_m * pzm + rhs;
        const float den = ex_c + exm + ey_c + ey_m + ez_c + ez_m +
                          KAPPA02 * lam * (RES * RES);

#if USE_ASYNC_LDS
        __builtin_amdgcn_s_wait_asynccnt(0);
#endif
        __syncthreads();                       // x+1 line now visible to block
        const float p_xp = has_xp ? sbuf[nxt][tz + 1] : 0.0f;

        num += ex_c * p_xp;
        phi_out[pb + s] = num / den;

        p_xm = p_c;
        exm  = ex_c;
        s   += BOX * BOX;
        __syncthreads();   // all reads of sbuf[cur] done before step x+1 DMA
    }
}

// ---------------------------------------------------------------------------
// launch: d_out = [ q (B*BOX3) | eps (B*4*BOX3) | phi (B*BOX3) ]
// d_ws   = [ phiA (B*BOX3) | phiB (B*BOX3) ]
// ---------------------------------------------------------------------------
extern "C" void kernel_launch(void* const* d_in, const int* in_sizes, int n_in,
                              void* d_out, int out_size, void* d_ws, size_t ws_size,
                              hipStream_t stream)
{
    (void)n_in; (void)out_size; (void)ws_size;
    const float* coords    = (const float*)d_in[0];
    const float* params    = (const float*)d_in[1];
    const int*   num_atoms = (const int*)d_in[2];

    const int B = in_sizes[2];
    const int N = in_sizes[1] / (2 * B);

    float* q_out   = (float*)d_out;                     // B * BOX3
    float* eps_out = q_out + (size_t)B * BOX3;          // B * 4 * BOX3
    float* phi_out = eps_out + (size_t)B * 4 * BOX3;    // B * BOX3

    float* phiA = (float*)d_ws;
    float* phiB = phiA + (size_t)B * BOX3;

    // zero q + eps accumulators (contiguous) and the initial phi
    (void)hipMemsetAsync(q_out, 0, (size_t)B * 5 * BOX3 * sizeof(float), stream);
    (void)hipMemsetAsync(phiA, 0, (size_t)B * BOX3 * sizeof(float), stream);

    // 1) eps log-occupancy accumulation: grid (N, 4, B)
    eps_accum_kernel<<<dim3(N, 4, B), 256, 0, stream>>>(
        coords, params, num_atoms, eps_out, N);

    // 2) charge deposition: grid (ceil(N/256), B)
    q_scatter_kernel<<<dim3((N + 255) / 256, B), 256, 0, stream>>>(
        coords, params, num_atoms, q_out, N);

    // 3) eps finalize (in place, float4)
    {
        const long long n4 = (long long)B * 4 * (BOX3 / 4);
        eps_finalize_kernel<<<(unsigned)((n4 + 255) / 256), 256, 0, stream>>>(
            (float4*)eps_out, n4);
    }

    // 4) 30 Jacobi sweeps, ping-pong in workspace, last one writes d_out.phi
    const float* src = phiA;
    float* bufs[2] = { phiB, phiA };
    for (int it = 0; it < NITER; ++it) {
        float* dst = (it == NITER - 1) ? phi_out : bufs[it & 1];
        jacobi_kernel<<<dim3(BOX / XCH, BOX, B), BOX, 0, stream>>>(
            src, dst, eps_out, q_out);
        src = dst;
    }
}